// GCNNet2_38500086841689
// MI455X (gfx1250) — compile-verified
//
#include <hip/hip_runtime.h>

typedef float v2f __attribute__((ext_vector_type(2)));
typedef float v8f __attribute__((ext_vector_type(8)));

#define NODES_D 128

// ---------------------------------------------------------------------------
// GEMM: out[N,128] = H[N,128] @ W[128,128] (+ bias, optional)
// Block = 256 threads = 8 wave32s. W staged in LDS pre-packed into WMMA
// B-fragment layout: frag f = (k0/4)*8 + j holds, per lane, the two floats
// {W[k][col], W[k+1][col]} the lane feeds to V_WMMA_F32_16X16X4_F32.
// Fragment read = one conflict-free ds_load_b64 per WMMA.
// A fragments are 2-stage prefetched global_load_b64.
// ---------------------------------------------------------------------------
__global__ void __launch_bounds__(256)
gemm_nn_128(const float* __restrict__ H, const float* __restrict__ W,
            const float* __restrict__ bias, float* __restrict__ out, int nRows)
{
    __shared__ v2f sB[8192];                 // 64KB: 256 frags x 32 lanes x v2f
    float* sBf = (float*)sB;

    const int tid = threadIdx.x;
    // cooperative W load, coalesced read, scatter into fragment layout
    #pragma unroll
    for (int it = 0; it < (NODES_D * NODES_D) / 256; ++it) {
        const int idx = tid + it * 256;
        const int k = idx >> 7;              // 0..127
        const int n = idx & 127;
        const int f = (k >> 2) * 8 + (n >> 4);
        const int laneSel = (n & 15) | (((k >> 1) & 1) << 4);
        const int slot = k & 1;
        sBf[f * 64 + laneSel * 2 + slot] = W[idx];
    }
    __syncthreads();

    const int wave = tid >> 5;               // 0..7
    const int lane = tid & 31;
    const int m    = lane & 15;
    const int koff = (lane >> 4) * 2;        // 0 (lanes 0-15) or 2 (lanes 16-31)

    const int row0 = blockIdx.x * 128 + wave * 16;
    const int row  = row0 + m;
    const int rowC = (row < nRows) ? row : (nRows - 1);   // clamp for safe loads
    const float* __restrict__ hrow = H + (size_t)rowC * NODES_D;

    v8f acc[8];
    #pragma unroll
    for (int j = 0; j < 8; ++j) acc[j] = (v8f){};

    v2f a_cur = *(const v2f*)(hrow + koff);
    #pragma unroll 2
    for (int k0 = 0; k0 < NODES_D; k0 += 4) {
        v2f a_nxt = a_cur;
        if (k0 + 4 < NODES_D)                // prefetch next A fragment
            a_nxt = *(const v2f*)(hrow + k0 + 4 + koff);
        const v2f* frag = sB + ((k0 >> 2) * 8) * 32 + lane;
        #pragma unroll
        for (int j = 0; j < 8; ++j) {
            v2f b = frag[j * 32];            // ds_load_b64, conflict-free
            acc[j] = __builtin_amdgcn_wmma_f32_16x16x4_f32(
                false, a_cur, false, b, (short)0, acc[j], false, false);
        }
        a_cur = a_nxt;
    }

    // preload the 8 bias values this lane needs (cols j*16+m), one branch
    float bv[8];
    #pragma unroll
    for (int j = 0; j < 8; ++j) bv[j] = 0.0f;
    if (bias) {
        #pragma unroll
        for (int j = 0; j < 8; ++j) bv[j] = bias[j * 16 + m];
    }

    // C/D layout: VGPR r -> lanes 0-15: M=r, N=lane ; lanes 16-31: M=8+r, N=lane-16
    const int mbase = (lane >> 4) * 8;
    #pragma unroll
    for (int r = 0; r < 8; ++r) {
        const int outRow = row0 + mbase + r;
        if (outRow < nRows) {
            float* orow = out + (size_t)outRow * NODES_D;
            #pragma unroll
            for (int j = 0; j < 8; ++j)
                orow[j * 16 + m] = acc[j][r] + bv[j];
        }
    }
}

// ---------------------------------------------------------------------------
// CSR-by-dst construction helpers
// ---------------------------------------------------------------------------
__global__ void __launch_bounds__(256)
zero_int(int* __restrict__ p, int n)
{
    int i = blockIdx.x * 256 + threadIdx.x;
    int stride = gridDim.x * 256;
    for (; i < n; i += stride) p[i] = 0;
}

__global__ void __launch_bounds__(256)
hist_deg(const int* __restrict__ dst, int* __restrict__ deg, int nEdges)
{
    const int e = blockIdx.x * 256 + threadIdx.x;
    if (e < nEdges) atomicAdd(&deg[dst[e]], 1);
}

// per-block inclusive scan of deg -> incl; block totals -> blockSums
__global__ void __launch_bounds__(256)
scan_block(const int* __restrict__ deg, int* __restrict__ incl,
           int* __restrict__ blockSums, int n)
{
    __shared__ int tmp[256];
    const int tid = threadIdx.x;
    const int i = blockIdx.x * 256 + tid;
    tmp[tid] = (i < n) ? deg[i] : 0;
    __syncthreads();
    #pragma unroll
    for (int off = 1; off < 256; off <<= 1) {
        int t = (tid >= off) ? tmp[tid - off] : 0;
        __syncthreads();
        tmp[tid] += t;
        __syncthreads();
    }
    if (i < n) incl[i] = tmp[tid];
    if (tid == 255) blockSums[blockIdx.x] = tmp[255];
}

// exclusive scan of block sums (small: ~196 entries), serial by thread 0
__global__ void scan_sums(int* __restrict__ blockSums, int nb)
{
    if (threadIdx.x == 0 && blockIdx.x == 0) {
        int acc = 0;
        for (int b = 0; b < nb; ++b) { int t = blockSums[b]; blockSums[b] = acc; acc += t; }
    }
}

// rowptr[i] = cursor[i] = exclusive prefix sum of deg
__global__ void __launch_bounds__(256)
finalize_rowptr(const int* __restrict__ deg, const int* __restrict__ incl,
                const int* __restrict__ blockSums, int* __restrict__ rowptr,
                int* __restrict__ cursor, int n)
{
    const int i = blockIdx.x * 256 + threadIdx.x;
    if (i < n) {
        const int excl = blockSums[blockIdx.x] + incl[i] - deg[i];
        rowptr[i] = excl;
        cursor[i] = excl;
    }
}

__global__ void __launch_bounds__(256)
fill_csr(const int* __restrict__ src, const int* __restrict__ dst,
         int* __restrict__ cursor, int* __restrict__ csr_src, int nEdges)
{
    const int e = blockIdx.x * 256 + threadIdx.x;
    if (e < nEdges) {
        const int pos = atomicAdd(&cursor[dst[e]], 1);
        csr_src[pos] = src[e];
    }
}

// ---------------------------------------------------------------------------
// Gather-aggregate (atomic-free): one wave per dst node, lanes hold float4 of
// the 128-wide row. out[n*stride + colOff + d] = relu(sum_in(lin[src]) + bias)
// node is forced wave-uniform so rowptr/deg/edge-list reads become s_load;
// lin (25.6MB) is L2-resident -> gather reads hit L2.
// ---------------------------------------------------------------------------
__global__ void __launch_bounds__(256)
gather_bias_relu(const float* __restrict__ lin, const int* __restrict__ csr_src,
                 const int* __restrict__ rowptr, const int* __restrict__ deg,
                 const float* __restrict__ bias, float* __restrict__ out,
                 int nRows, int outStride, int colOff)
{
    const int node = __builtin_amdgcn_readfirstlane(blockIdx.x * 8 + (threadIdx.x >> 5));
    if (node >= nRows) return;
    const int lane = threadIdx.x & 31;
    const int* __restrict__ ep = csr_src + rowptr[node];
    const int cnt = deg[node];

    float4 acc; acc.x = acc.y = acc.z = acc.w = 0.0f;
    int i = 0;
    for (; i + 1 < cnt; i += 2) {
        const int s0 = ep[i], s1 = ep[i + 1];
        const float4 v0 = ((const float4*)(lin + (size_t)s0 * NODES_D))[lane];
        const float4 v1 = ((const float4*)(lin + (size_t)s1 * NODES_D))[lane];
        acc.x += v0.x + v1.x; acc.y += v0.y + v1.y;
        acc.z += v0.z + v1.z; acc.w += v0.w + v1.w;
    }
    if (i < cnt) {
        const float4 v = ((const float4*)(lin + (size_t)ep[i] * NODES_D))[lane];
        acc.x += v.x; acc.y += v.y; acc.z += v.z; acc.w += v.w;
    }
    const float4 b4 = ((const float4*)bias)[lane];
    float4 r;
    r.x = fmaxf(acc.x + b4.x, 0.0f);
    r.y = fmaxf(acc.y + b4.y, 0.0f);
    r.z = fmaxf(acc.z + b4.z, 0.0f);
    r.w = fmaxf(acc.w + b4.w, 0.0f);
    *(float4*)(out + (size_t)node * outStride + colOff + lane * 4) = r;
}

// ---------------------------------------------------------------------------
// Readout: logits[n] = feat[n,:384] . rw ; sig = sigmoid(logits). Wave/node.
// ---------------------------------------------------------------------------
__global__ void __launch_bounds__(256)
readout_384(const float* __restrict__ feat, const float* __restrict__ rw,
            float* __restrict__ logits, float* __restrict__ sig, int nRows)
{
    const int node = __builtin_amdgcn_readfirstlane(blockIdx.x * 8 + (threadIdx.x >> 5));
    if (node >= nRows) return;
    const int lane = threadIdx.x & 31;
    const float4* fr = (const float4*)(feat + (size_t)node * 384);
    const float4* wr = (const float4*)rw;
    float s = 0.0f;
    #pragma unroll
    for (int i = 0; i < 3; ++i) {
        float4 f = fr[lane + 32 * i];
        float4 w = wr[lane + 32 * i];
        s += f.x * w.x + f.y * w.y + f.z * w.z + f.w * w.w;
    }
    #pragma unroll
    for (int off = 16; off > 0; off >>= 1) s += __shfl_xor(s, off, 32);
    if (lane == 0) {
        logits[node] = s;
        sig[node] = 1.0f / (1.0f + expf(-s));
    }
}

// ---------------------------------------------------------------------------
extern "C" void kernel_launch(void* const* d_in, const int* in_sizes, int n_in,
                              void* d_out, int out_size, void* d_ws, size_t ws_size,
                              hipStream_t stream)
{
    const float* x      = (const float*)d_in[0];
    const int*   edge   = (const int*)  d_in[1];
    const float* emb_W  = (const float*)d_in[2];
    const float* emb_b  = (const float*)d_in[3];
    const float* gcn_W  = (const float*)d_in[4];   // [4,128,128] (layer 3 is dead)
    const float* gcn_b  = (const float*)d_in[5];   // [4,128]
    const float* skip_W = (const float*)d_in[6];   // [3,128,128]
    const float* skip_b = (const float*)d_in[7];   // [3,128]
    const float* rw     = (const float*)d_in[8];   // [384]

    const int N = in_sizes[0] / NODES_D;
    const int E = in_sizes[1] / 2;
    const int* src = edge;
    const int* dst = edge + E;

    // workspace: 5 feature buffers + CSR ints
    float* f = (float*)d_ws;
    const size_t NB = (size_t)N * NODES_D;
    float* h1  = f;
    float* h2  = f + 1 * NB;
    float* h3  = f + 2 * NB;
    float* lin = f + 3 * NB;
    float* h0  = f + 4 * NB;
    int* ip        = (int*)(f + 5 * NB);
    int* deg       = ip;
    int* rowptr    = ip + N;
    int* cursor    = ip + 2 * N;        // also scratch for inclusive scan
    int* blockSums = ip + 3 * N;
    int* csr_src   = ip + 3 * N + 4096;

    float* out_feat = (float*)d_out;               // [N,384]
    float* logits   = out_feat + (size_t)N * 384;  // [N]
    float* sigv     = logits + N;                  // [N]

    const dim3 blk(256);
    const dim3 gGemm((N + 127) / 128);
    const dim3 gEdge((E + 255) / 256);
    const int  nScan = (N + 255) / 256;
    const dim3 gScan(nScan);
    const dim3 gNode((N + 7) / 8);

    // ---- Build dst-CSR once (amortized over 6 propagations) ----
    zero_int<<<64, blk, 0, stream>>>(deg, N);
    hist_deg<<<gEdge, blk, 0, stream>>>(dst, deg, E);
    scan_block<<<gScan, blk, 0, stream>>>(deg, cursor, blockSums, N);
    scan_sums<<<1, 32, 0, stream>>>(blockSums, nScan);
    finalize_rowptr<<<gScan, blk, 0, stream>>>(deg, cursor, blockSums, rowptr, cursor, N);
    fill_csr<<<gEdge, blk, 0, stream>>>(src, dst, cursor, csr_src, E);

    // ---- Embedding: h0 = x @ emb_W + emb_b ----
    gemm_nn_128<<<gGemm, blk, 0, stream>>>(x, emb_W, emb_b, h0, N);

    float* feats[4] = { h0, h1, h2, h3 };

    // ---- Three live GCN layers (layer 3 output is never consumed) ----
    for (int i = 0; i < 3; ++i) {
        gemm_nn_128<<<gGemm, blk, 0, stream>>>(feats[i], gcn_W + (size_t)i * NODES_D * NODES_D,
                                               nullptr, lin, N);
        gather_bias_relu<<<gNode, blk, 0, stream>>>(lin, csr_src, rowptr, deg,
                                                    gcn_b + (size_t)i * NODES_D,
                                                    feats[i + 1], N, NODES_D, 0);
    }

    // ---- Skip convs on feats[1..3] -> strided into d_out[:, j*128:(j+1)*128] ----
    for (int j = 0; j < 3; ++j) {
        gemm_nn_128<<<gGemm, blk, 0, stream>>>(feats[j + 1], skip_W + (size_t)j * NODES_D * NODES_D,
                                               nullptr, lin, N);
        gather_bias_relu<<<gNode, blk, 0, stream>>>(lin, csr_src, rowptr, deg,
                                                    skip_b + (size_t)j * NODES_D,
                                                    out_feat, N, 384, j * NODES_D);
    }

    // ---- Readout + sigmoid ----
    readout_384<<<gNode, blk, 0, stream>>>(out_feat, rw, logits, sigv, N);
}